// QuantumLayer_61375082660337
// MI455X (gfx1250) — compile-verified
//
#include <hip/hip_runtime.h>
#include <math.h>

// QuantumLayer: 12-qubit state-vector sim, batched.
// One 256-thread block per batch row; state (re/im f32, 4096 each) lives in LDS.
// Per layer: three 16x16 complex group-unitaries applied via V_WMMA_F32_16X16X4_F32,
// then the CNOT ring as a composed bit-permutation gather.

typedef __attribute__((ext_vector_type(2))) float v2f;
typedef __attribute__((ext_vector_type(8))) float v8f;

#define NQ 12
#define DIM 4096
#define NLAYERS 4
#define TPB 256
#define NWAVES (TPB / 32)
#define ROWS_PER_THREAD (DIM / TPB) // 16

// qubit q <-> flat bit (11-q).  Group g covers qubits 4g..4g+3.
// flat index for group sub-index k (4 bits) and column col (8 remaining bits):
__device__ __forceinline__ int flat_idx(int g, int k, int col) {
  if (g == 0) return (k << 8) | col;                                     // bits 11..8
  else if (g == 1) return ((col & 0xF0) << 4) | (k << 4) | (col & 0x0F); // bits 7..4
  else return (col << 4) | k;                                            // bits 3..0
}

// Apply 16x16 complex gate (Gr + i*Gi, row-major 16x16 in LDS) along group-g axis.
// D = G * S over 256 columns, in place.  8 waves x 2 column-tiles of 16.
__device__ __forceinline__ void apply_gate(int g,
                                           const float* __restrict__ Gr,
                                           const float* __restrict__ Gi,
                                           float* s_re, float* s_im,
                                           int lane, int wid) {
  const int m  = lane & 15;   // A-matrix row / B,C,D column
  const int hi = lane >> 4;   // lane half selects K offset (A/B) and M offset (C/D)

  // A fragments for all 4 K-steps: V_WMMA_F32_16X16X4_F32 layout:
  // lane L: M = L&15; vgpr0 = A[M][kb], vgpr1 = A[M][kb+1], kb = (L<16 ? 0 : 2).
  v2f Ar[4], Ai[4], An[4];
#pragma unroll
  for (int ks = 0; ks < 4; ++ks) {
    const int kb = ks * 4 + hi * 2;
    v2f ar, ai;
    ar.x = Gr[m * 16 + kb]; ar.y = Gr[m * 16 + kb + 1];
    ai.x = Gi[m * 16 + kb]; ai.y = Gi[m * 16 + kb + 1];
    Ar[ks] = ar; Ai[ks] = ai; An[ks] = -ai; // f32 WMMA has no A-neg -> pre-negate
  }

  // Fixed trip count of 2 (uniform, fully unrolled): no EXEC-masked loop around WMMA.
#pragma unroll
  for (int tt = 0; tt < 2; ++tt) {
    const int t   = wid + tt * 8;
    const int col = t * 16 + m;
    v2f Br[4], Bi[4];
#pragma unroll
    for (int ks = 0; ks < 4; ++ks) {
      const int k0 = ks * 4 + hi * 2;
      const int f0 = flat_idx(g, k0, col);
      const int f1 = flat_idx(g, k0 + 1, col);
      v2f br, bi;
      br.x = s_re[f0]; br.y = s_re[f1];
      bi.x = s_im[f0]; bi.y = s_im[f1];
      Br[ks] = br; Bi[ks] = bi;
    }
    v8f cre = {0.f, 0.f, 0.f, 0.f, 0.f, 0.f, 0.f, 0.f};
    v8f cim = {0.f, 0.f, 0.f, 0.f, 0.f, 0.f, 0.f, 0.f};
#pragma unroll
    for (int ks = 0; ks < 4; ++ks) {
      // Re(D) = Gr*Sr - Gi*Si ; Im(D) = Gr*Si + Gi*Sr
      cre = __builtin_amdgcn_wmma_f32_16x16x4_f32(false, Ar[ks], false, Br[ks], (short)0, cre, false, false);
      cre = __builtin_amdgcn_wmma_f32_16x16x4_f32(false, An[ks], false, Bi[ks], (short)0, cre, false, false);
      cim = __builtin_amdgcn_wmma_f32_16x16x4_f32(false, Ar[ks], false, Bi[ks], (short)0, cim, false, false);
      cim = __builtin_amdgcn_wmma_f32_16x16x4_f32(false, Ai[ks], false, Br[ks], (short)0, cim, false, false);
    }
    // D layout: vgpr v, lane L -> row = v + 8*(L>>4), col = L&15
#pragma unroll
    for (int v = 0; v < 8; ++v) {
      const int mm = v + hi * 8;
      const int f  = flat_idx(g, mm, col);
      s_re[f] = cre[v];
      s_im[f] = cim[v];
    }
  }
}

__global__ __launch_bounds__(TPB) void quantum_layer_kernel(
    const float* __restrict__ x, const float* __restrict__ w,
    float* __restrict__ out) {
  __shared__ float s_re[DIM];
  __shared__ float s_im[DIM];
  __shared__ float g_re[NLAYERS * 3 * 256];
  __shared__ float g_im[NLAYERS * 3 * 256];
  __shared__ float red[TPB];
  __shared__ float wred[NWAVES * NQ];
  __shared__ float wsh[NLAYERS * NQ * 2];

  const int tid  = threadIdx.x;
  const int lane = tid & 31;
  const int wid  = tid >> 5;

  if (tid < NLAYERS * NQ * 2) wsh[tid] = w[tid];

  // ---- Load row (float4-coalesced), accumulate norm ----
  const float4* xb4 = (const float4*)(x + (size_t)blockIdx.x * DIM);
  float ss = 0.f;
#pragma unroll
  for (int r = 0; r < 4; ++r) {
    const int e = r * TPB + tid;       // float4 element index
    const float4 v = xb4[e];
    s_re[e * 4 + 0] = v.x; s_re[e * 4 + 1] = v.y;
    s_re[e * 4 + 2] = v.z; s_re[e * 4 + 3] = v.w;
    s_im[e * 4 + 0] = 0.f; s_im[e * 4 + 1] = 0.f;
    s_im[e * 4 + 2] = 0.f; s_im[e * 4 + 3] = 0.f;
    ss += v.x * v.x + v.y * v.y + v.z * v.z + v.w * v.w;
  }
  // wave-level reduce, then one cross-wave combine
#pragma unroll
  for (int off = 16; off > 0; off >>= 1) ss += __shfl_down(ss, off, 32);
  if (lane == 0) red[wid] = ss;
  __syncthreads();
  float tot = 0.f;
#pragma unroll
  for (int i = 0; i < NWAVES; ++i) tot += red[i];
  const float rn = rsqrtf(tot);
  // scale (same indexing as the write above -> same-thread elements, no barrier needed)
#pragma unroll
  for (int r = 0; r < 4; ++r) {
    const int e = r * TPB + tid;
    s_re[e * 4 + 0] *= rn; s_re[e * 4 + 1] *= rn;
    s_re[e * 4 + 2] *= rn; s_re[e * 4 + 3] *= rn;
  }

  // ---- Build 16x16 group unitaries for all layers ----
  // U_q = RZ(tz) RY(ty) = [[c*em, -s*em],[s*ep, c*ep]], em = cz - i sz, ep = cz + i sz
  // group matrix entry [a][b] = prod_j U_{4g+j}[a_{3-j}][b_{3-j}]  (qubit 4g = MSB of a)
  for (int it = 0; it < NLAYERS * 3; ++it) {
    const int l = it / 3, g = it % 3;
    const int a = tid >> 4, b = tid & 15;
    float re = 1.f, im = 0.f;
#pragma unroll
    for (int j = 0; j < 4; ++j) {
      const int q  = g * 4 + j;
      const int ra = (a >> (3 - j)) & 1;
      const int rb = (b >> (3 - j)) & 1;
      const float ty = wsh[(l * NQ + q) * 2 + 0];
      const float tz = wsh[(l * NQ + q) * 2 + 1];
      float sy, cy, sz, cz;
      __sincosf(0.5f * ty, &sy, &cy);
      __sincosf(0.5f * tz, &sz, &cz);
      const float amp = (ra == 0) ? ((rb == 0) ? cy : -sy) : ((rb == 0) ? sy : cy);
      const float ure = amp * cz;
      const float uim = (ra == 0) ? (-amp * sz) : (amp * sz);
      const float nre = re * ure - im * uim;
      const float nim = re * uim + im * ure;
      re = nre; im = nim;
    }
    g_re[it * 256 + tid] = re;
    g_im[it * 256 + tid] = im;
  }
  __syncthreads();

  // ---- Circuit ----
  for (int l = 0; l < NLAYERS; ++l) {
#pragma unroll
    for (int g = 0; g < 3; ++g) {
      apply_gate(g, &g_re[(l * 3 + g) * 256], &g_im[(l * 3 + g) * 256],
                 s_re, s_im, lane, wid);
      __syncthreads();
    }
    // CNOT ring composed into one permutation: new[i] = old[src(i)],
    // src = sigma_1(...sigma_12(i)), applied last-gate-first.
    float tr[ROWS_PER_THREAD], ti[ROWS_PER_THREAD];
#pragma unroll
    for (int r = 0; r < ROWS_PER_THREAD; ++r) {
      const int i = r * TPB + tid;
      unsigned jj = (unsigned)i;
      jj ^= (jj & 1u) << 11;                        // CNOT(11,0): c=bit0, t=bit11
#pragma unroll
      for (int k = 11; k >= 1; --k)                 // CNOT(k-1,k): c=bit(12-k), t=bit(11-k)
        jj ^= ((jj >> (12 - k)) & 1u) << (11 - k);
      tr[r] = s_re[jj];
      ti[r] = s_im[jj];
    }
    __syncthreads();
#pragma unroll
    for (int r = 0; r < ROWS_PER_THREAD; ++r) {
      const int i = r * TPB + tid;
      s_re[i] = tr[r];
      s_im[i] = ti[r];
    }
    __syncthreads();
  }

  // ---- <Z_q> expectation values: per-thread partials, wave shuffle-reduce,
  //      then one cross-wave combine with a single barrier ----
  float part[NQ];
#pragma unroll
  for (int q = 0; q < NQ; ++q) part[q] = 0.f;
#pragma unroll
  for (int r = 0; r < ROWS_PER_THREAD; ++r) {
    const int i = r * TPB + tid;
    const float pre = s_re[i], pim = s_im[i];
    const float p = pre * pre + pim * pim;
#pragma unroll
    for (int q = 0; q < NQ; ++q)
      part[q] += ((i >> (11 - q)) & 1) ? -p : p;
  }
#pragma unroll
  for (int q = 0; q < NQ; ++q) {
#pragma unroll
    for (int off = 16; off > 0; off >>= 1)
      part[q] += __shfl_down(part[q], off, 32);
  }
  if (lane < NQ) wred[wid * NQ + lane] = part[lane]; // lane0 holds the wave sums... 
  // NOTE: after shfl_down reduction the total sits in lane 0 for every q; broadcast:
  // simpler & correct: lane 0 writes all NQ sums.
  if (lane == 0) {
#pragma unroll
    for (int q = 0; q < NQ; ++q) wred[wid * NQ + q] = part[q];
  }
  __syncthreads();
  if (tid < NQ) {
    float acc = 0.f;
#pragma unroll
    for (int i = 0; i < NWAVES; ++i) acc += wred[i * NQ + tid];
    out[(size_t)blockIdx.x * NQ + tid] = acc;
  }
}

extern "C" void kernel_launch(void* const* d_in, const int* in_sizes, int n_in,
                              void* d_out, int out_size, void* d_ws, size_t ws_size,
                              hipStream_t stream) {
  const float* x = (const float*)d_in[0];     // (BATCH, 4096) f32
  const float* w = (const float*)d_in[1];     // (4, 12, 2) f32
  float* out = (float*)d_out;                 // (BATCH, 12) f32
  const int batch = in_sizes[0] / DIM;        // 4096
  quantum_layer_kernel<<<dim3(batch), dim3(TPB), 0, stream>>>(x, w, out);
}